// RXGate_549755814187
// MI455X (gfx1250) — compile-verified
//
#include <hip/hip_runtime.h>

// CDNA5 / gfx1250: wave32, WMMA f32 16x16x4, async global<->LDS staging.
typedef __attribute__((ext_vector_type(2))) float v2f;
typedef __attribute__((ext_vector_type(8))) float v8f;

// Builtin-matching pointer types for global_load_async_to_lds_b128:
// param0: 16-byte int vector in global (AS1), param1: same in LDS (AS3).
typedef int v4si __attribute__((vector_size(16)));
typedef __attribute__((address_space(1))) v4si v4si_g;
typedef __attribute__((address_space(3))) v4si v4si_l;

#define DIMS 8192   // 2^13 state dim
#define BCOL 64     // batch columns
#define LS   68     // padded LDS row stride (floats); 68*4=272 bytes, 16B-aligned rows

// ---- async global <-> LDS (ASYNCcnt path) -------------------------------
__device__ __forceinline__ void async_load_b128(const void* gsrc, const void* ldst) {
  __builtin_amdgcn_global_load_async_to_lds_b128(
      (v4si_g*)(unsigned long long)gsrc,
      (v4si_l*)(unsigned int)(unsigned long long)ldst,
      0, 0);
}
__device__ __forceinline__ void async_store_b128(void* gdst, const void* lsrc) {
  __builtin_amdgcn_global_store_async_from_lds_b128(
      (v4si_g*)(unsigned long long)gdst,
      (v4si_l*)(unsigned int)(unsigned long long)lsrc,
      0, 0);
}

// ---- WMMA core ----------------------------------------------------------
__device__ __forceinline__ v8f wmma_f32(v2f a, v2f b, v8f c) {
  // D = A(16x4) * B(4x16) + C(16x16), f32 in/out.
  return __builtin_amdgcn_wmma_f32_16x16x4_f32(false, a, false, b, (short)0, c, false, false);
}

// Row mapping inside a 16-element transform group, per phase.
template <int PH>
__device__ __forceinline__ int rowmap(int r) {
  if (PH == 2) return (r & 1) + ((r >> 1) << 4);
  if (PH == 4) return (r & 3) + ((r >> 2) << 4);
  return r;
}

// A 16x4 layout: lanes 0-15 hold M=lane, VGPR0=K0,VGPR1=K1; lanes 16-31 hold K2,K3.
__device__ __forceinline__ void load_wfrags(const float* __restrict__ wre,
                                            const float* __restrict__ wim,
                                            v2f* are, v2f* aim, v2f* aimn) {
  const int lane = threadIdx.x & 31;
  const int row  = lane & 15;
  const int kof  = (lane >> 4) << 1; // 0 or 2
#pragma unroll
  for (int k = 0; k < 4; ++k) {
    const int c0 = 4 * k + kof;
    v2f ar; ar.x = wre[row * 16 + c0]; ar.y = wre[row * 16 + c0 + 1];
    v2f ai; ai.x = wim[row * 16 + c0]; ai.y = wim[row * 16 + c0 + 1];
    are[k]  = ar;
    aim[k]  = ai;
    aimn[k] = -ai; // f32 WMMA has no A-negate; pre-negate for the -Wim*Xim term
  }
}

// One complex 16x16 * 16x16 tile, in place in LDS.
template <int PH>
__device__ __forceinline__ void do_tile(float* __restrict__ sre, float* __restrict__ sim,
                                        int base, int n0,
                                        const v2f* are, const v2f* aim, const v2f* aimn) {
  const int lane  = threadIdx.x & 31;
  const int l     = lane & 15;
  const int half2 = (lane >> 4) << 1; // 0 or 2
  v2f bre[4], bim[4];
#pragma unroll
  for (int k = 0; k < 4; ++k) {
    // B 4x16 layout: lanes 0-15: N=lane, VGPR0=K0,VGPR1=K1; lanes 16-31: K2,K3.
    const int c0 = 4 * k + half2;
    const int i0 = (base + rowmap<PH>(c0))     * LS + n0 + l;
    const int i1 = (base + rowmap<PH>(c0 + 1)) * LS + n0 + l;
    v2f br; br.x = sre[i0]; br.y = sre[i1];
    v2f bi; bi.x = sim[i0]; bi.y = sim[i1];
    bre[k] = br; bim[k] = bi;
  }
  v8f cre = {}; v8f cim = {};
#pragma unroll
  for (int k = 0; k < 4; ++k) {
    cre = wmma_f32(are[k],  bre[k], cre);  // + Wre*Xre
    cre = wmma_f32(aimn[k], bim[k], cre);  // - Wim*Xim
    cim = wmma_f32(are[k],  bim[k], cim);  // + Wre*Xim
    cim = wmma_f32(aim[k],  bre[k], cim);  // + Wim*Xre
  }
  // D 16x16 f32 layout: lanes 0-15: N=lane, elem v -> M=v; lanes 16-31: M=v+8.
#pragma unroll
  for (int v = 0; v < 8; ++v) {
    const int m   = v + (half2 << 2); // +0 or +8
    const int idx = (base + rowmap<PH>(m)) * LS + n0 + l;
    sre[idx] = cre[v];
    sim[idx] = cim[v];
  }
}

// Build the four 16x16 kron-gate matrices (re,im) from angle[13], CUTOFF=10 Taylor.
__global__ void rx_prep(const float* __restrict__ angle, float* __restrict__ w) {
  float av[13], bv[13];
#pragma unroll
  for (int g = 0; g < 13; ++g) {
    const float t  = 0.5f * angle[g];
    const float t2 = t * t;
    av[g] = 1.f + t2 * (-0.5f + t2 * ((1.f / 24.f) + t2 * ((-1.f / 720.f) +
              t2 * ((1.f / 40320.f) + t2 * (-1.f / 3628800.f)))));
    bv[g] = -t * (1.f + t2 * ((-1.f / 6.f) + t2 * ((1.f / 120.f) +
              t2 * ((-1.f / 5040.f) + t2 * (1.f / 362880.f)))));
  }
  const int pos[4][4] = {{0, 1, 2, 3}, {0, 4, 5, 6}, {7, 8, 9, 10}, {7, 8, 11, 12}};
  const int idm[4]    = {0x0, 0x1, 0x0, 0x3}; // identity-slot masks
  const int r = threadIdx.x >> 4, c = threadIdx.x & 15;
#pragma unroll
  for (int grp = 0; grp < 4; ++grp) {
    float val = 1.f; int d = 0; bool zero = false;
#pragma unroll
    for (int k = 0; k < 4; ++k) {
      const int  rk = (r >> k) & 1, ck = (c >> k) & 1;
      const bool id = (idm[grp] >> k) & 1;
      const int  g  = 12 - pos[grp][k];
      if (rk == ck) { val *= id ? 1.f : av[g]; }
      else          { if (id) zero = true; else { val *= bv[g]; ++d; } }
    }
    float re = 0.f, im = 0.f;
    if (!zero) {
      switch (d & 3) {
        case 0: re =  val; break;
        case 1: im =  val; break;
        case 2: re = -val; break;
        case 3: im = -val; break;
      }
    }
    w[grp * 512 +       r * 16 + c] = re;
    w[grp * 512 + 256 + r * 16 + c] = im;
  }
}

// Pass 1: 64 WGs x 128 contiguous rows x 64 cols. Applies bit groups {0-3}, {0,4,5,6}.
__global__ void __launch_bounds__(256) rx_pass1(const float* __restrict__ xre,
                                                const float* __restrict__ xim,
                                                const float* __restrict__ w,
                                                float* __restrict__ ore,
                                                float* __restrict__ oim) {
  __shared__ float sre[128 * LS];
  __shared__ float sim[128 * LS];
  const int blk = blockIdx.x;
  const int t   = threadIdx.x;
  const float* gre = xre + (size_t)blk * (128 * BCOL);
  const float* gim = xim + (size_t)blk * (128 * BCOL);
  // Async fill: 2048 16-byte chunks per component (128 rows x 16 chunks).
  for (int j = t; j < 2048; j += 256) {
    const int rr = j >> 4, c4 = j & 15;
    async_load_b128(gre + j * 4, &sre[rr * LS + c4 * 4]);
    async_load_b128(gim + j * 4, &sim[rr * LS + c4 * 4]);
  }
  __builtin_amdgcn_s_wait_asynccnt(0);
  __syncthreads();

  const int wave = t >> 5;
  v2f are[4], aim[4], aimn[4];
  load_wfrags(w, w + 256, are, aim, aimn);            // W1: bits 0-3 (gates 12..9)
  for (int tile = wave; tile < 32; tile += 8) {
    const int o = tile >> 2, n0 = (tile & 3) << 4;
    do_tile<1>(sre, sim, o * 16, n0, are, aim, aimn);
  }
  __syncthreads();
  load_wfrags(w + 512, w + 768, are, aim, aimn);      // W2: bits {0,4,5,6} (id,8,7,6)
  for (int tile = wave; tile < 32; tile += 8) {
    const int o = tile >> 2, n0 = (tile & 3) << 4;
    do_tile<2>(sre, sim, o * 2, n0, are, aim, aimn);
  }
  __syncthreads();
  // Async drain to the intermediate (d_out); ENDPGM's implicit wait-idle fences it.
  float* dre = ore + (size_t)blk * (128 * BCOL);
  float* dim_ = oim + (size_t)blk * (128 * BCOL);
  for (int j = t; j < 2048; j += 256) {
    const int rr = j >> 4, c4 = j & 15;
    async_store_b128(dre + j * 4, &sre[rr * LS + c4 * 4]);
    async_store_b128(dim_ + j * 4, &sim[rr * LS + c4 * 4]);
  }
}

// Pass 2: 128 WGs; WG `lo` owns rows lo + hi*128 (hi=0..63) x 64 cols, in place in d_out.
__global__ void __launch_bounds__(256) rx_pass2(float* __restrict__ ore,
                                                float* __restrict__ oim,
                                                const float* __restrict__ w) {
  __shared__ float sre[64 * LS];
  __shared__ float sim[64 * LS];
  const int lo = blockIdx.x;
  const int t  = threadIdx.x;
  // Async gather: 1024 16-byte chunks per component (64 strided rows x 16 chunks).
  for (int j = t; j < 1024; j += 256) {
    const int hi = j >> 4, c4 = j & 15;
    const size_t g = (size_t)(lo + hi * 128) * BCOL + c4 * 4;
    async_load_b128(ore + g, &sre[hi * LS + c4 * 4]);
    async_load_b128(oim + g, &sim[hi * LS + c4 * 4]);
  }
  __builtin_amdgcn_s_wait_asynccnt(0);
  __syncthreads();

  const int wave = t >> 5;
  v2f are[4], aim[4], aimn[4];
  load_wfrags(w + 1024, w + 1280, are, aim, aimn);    // W3: bits 7-10 (gates 5..2)
  for (int tile = wave; tile < 16; tile += 8) {
    const int o = tile >> 2, n0 = (tile & 3) << 4;
    do_tile<3>(sre, sim, o * 16, n0, are, aim, aimn);
  }
  __syncthreads();
  load_wfrags(w + 1536, w + 1792, are, aim, aimn);    // W4: bits {7,8,11,12} (id,id,1,0)
  for (int tile = wave; tile < 16; tile += 8) {
    const int o = tile >> 2, n0 = (tile & 3) << 4;
    do_tile<4>(sre, sim, o * 4, n0, are, aim, aimn);
  }
  __syncthreads();
  for (int j = t; j < 1024; j += 256) {
    const int hi = j >> 4, c4 = j & 15;
    const size_t g = (size_t)(lo + hi * 128) * BCOL + c4 * 4;
    async_store_b128(ore + g, &sre[hi * LS + c4 * 4]);
    async_store_b128(oim + g, &sim[hi * LS + c4 * 4]);
  }
}

extern "C" void kernel_launch(void* const* d_in, const int* in_sizes, int n_in,
                              void* d_out, int out_size, void* d_ws, size_t ws_size,
                              hipStream_t stream) {
  const float* xre   = (const float*)d_in[0]; // [8192*64]
  const float* xim   = (const float*)d_in[1]; // [8192*64]
  const float* angle = (const float*)d_in[2]; // [13]
  float* w   = (float*)d_ws;                  // 4 matrices x (256 re + 256 im) = 8 KB
  float* ore = (float*)d_out;                 // out[0] = real plane
  float* oim = ore + (size_t)DIMS * BCOL;     // out[1] = imag plane

  rx_prep <<<1,   256, 0, stream>>>(angle, w);
  rx_pass1<<<64,  256, 0, stream>>>(xre, xim, w, ore, oim);
  rx_pass2<<<128, 256, 0, stream>>>(ore, oim, w);
}